// NMS_46042049413158
// MI455X (gfx1250) — compile-verified
//
#include <hip/hip_runtime.h>
#include <cstdint>
#include <cstddef>

// ---------------- problem constants (match reference) ----------------
#define BATCH       4
#define NBOX        200000
#define NCLS        10
#define KSEL        4096          // NMS_PRE_MAXSIZE
#define POST        500           // NMS_POST_MAXSIZE
#define NMS_TH      0.5f
#define SCORE_TH    0.1f
#define NWORD       (KSEL / 32)   // 128 mask words per row (wave32 ballots)

typedef float v2f __attribute__((ext_vector_type(2)));
typedef float v8f __attribute__((ext_vector_type(8)));

// monotone float->uint mapping (larger float => larger uint); 0 reserved = invalid
__device__ __forceinline__ unsigned orderable(float f) {
  unsigned u = __float_as_uint(f);
  return (u & 0x80000000u) ? ~u : (u | 0x80000000u);
}
__device__ __forceinline__ float fromOrderable(unsigned u) {
  return (u & 0x80000000u) ? __uint_as_float(u & 0x7FFFFFFFu)
                           : __uint_as_float(~u);
}

// ---------------- k0: zero histogram ----------------
__global__ void k_zero(int* __restrict__ p, int n) {
  int i = blockIdx.x * blockDim.x + threadIdx.x;
  if (i < n) p[i] = 0;
}

// ---------------- k1: class-max + orderable key + 16-bit-bin histogram ----------------
__global__ void __launch_bounds__(256)
k_score(const float* __restrict__ cls, unsigned* __restrict__ uArr, int* __restrict__ hist) {
  int gid = blockIdx.x * blockDim.x + threadIdx.x;
  if (gid >= BATCH * NBOX) return;
  int b = gid / NBOX;
  const float* p = cls + (size_t)gid * NCLS;
  float m = p[0];
#pragma unroll
  for (int c = 1; c < NCLS; ++c) m = fmaxf(m, p[c]);
  unsigned u = (m > SCORE_TH) ? orderable(m) : 0u;
  uArr[gid] = u;
  if (u) atomicAdd(&hist[b * 65536 + (int)(u >> 16)], 1);
}

// ---------------- k2: find threshold bin for exact top-KSEL ----------------
// params[b] = { binT, cntAbove, need, pad }
__global__ void __launch_bounds__(1024)
k_select(const int* __restrict__ hist, int* __restrict__ params) {
  const int b = blockIdx.x;
  const int tid = threadIdx.x;
  __shared__ int cs[1024];
  const int* h = hist + b * 65536;
  int s = 0;
  for (int bin = tid * 64; bin < tid * 64 + 64; ++bin) s += h[bin];
  cs[tid] = s;
  __syncthreads();
  if (tid == 0) {
    int cum = 0, c;
    for (c = 1023; c >= 0; --c) {
      if (cum + cs[c] >= KSEL) break;
      cum += cs[c];
    }
    int binT, cntAbove;
    if (c < 0) {                       // fewer than KSEL valid entries
      binT = 0;
      cntAbove = cum - h[0];
    } else {
      binT = c * 64;
      for (int bin = c * 64 + 63; bin >= c * 64; --bin) {
        int hv = h[bin];
        if (cum + hv >= KSEL) { binT = bin; break; }
        cum += hv;
      }
      cntAbove = cum;
    }
    params[b * 4 + 0] = binT;
    params[b * 4 + 1] = cntAbove;
    params[b * 4 + 2] = KSEL - cntAbove;
  }
}

// ---------------- k3: deterministic ordered compaction into KSEL slots ----------------
__global__ void __launch_bounds__(1024)
k_compact(const unsigned* __restrict__ uArr, const int* __restrict__ params,
          unsigned* __restrict__ topU, int* __restrict__ topIdx) {
  const int b = blockIdx.x;
  const int tid = threadIdx.x;
  const int lane = tid & 31, wv = tid >> 5;
  const int binT = params[b * 4 + 0];
  const int cntAbove = params[b * 4 + 1];
  const int need = params[b * 4 + 2];
  __shared__ int wA[32], wE[32];
  __shared__ int baseA, baseE;
  if (tid == 0) { baseA = 0; baseE = 0; }
  __syncthreads();
  const unsigned laneLt = (1u << lane) - 1u;
  for (int base = 0; base < NBOX; base += 1024) {
    int i = base + tid;
    unsigned u = (i < NBOX) ? uArr[(size_t)b * NBOX + i] : 0u;
    int bin = (int)(u >> 16);
    bool pA = (u != 0u) && (bin > binT);
    bool pE = (u != 0u) && (bin == binT);
    unsigned mA = (unsigned)__ballot(pA);
    unsigned mE = (unsigned)__ballot(pE);
    if (lane == 0) { wA[wv] = __popc(mA); wE[wv] = __popc(mE); }
    int bA = baseA, bE = baseE;        // stable since previous end-barrier
    __syncthreads();
    int wpA = 0, wpE = 0, totA = 0, totE = 0;
    for (int w = 0; w < 32; ++w) {
      int a = wA[w], e = wE[w];
      if (w < wv) { wpA += a; wpE += e; }
      totA += a; totE += e;
    }
    if (pA) {
      int pos = bA + wpA + __popc(mA & laneLt);
      topU[b * KSEL + pos] = u; topIdx[b * KSEL + pos] = i;
    }
    if (pE) {
      int s = bE + wpE + __popc(mE & laneLt);
      if (s < need) {
        int pos = cntAbove + s;
        topU[b * KSEL + pos] = u; topIdx[b * KSEL + pos] = i;
      }
    }
    __syncthreads();
    if (tid == 0) { baseA = bA + totA; baseE = bE + totE; }
    __syncthreads();
  }
  int takenE = baseE < need ? baseE : need;
  for (int i = cntAbove + takenE + tid; i < KSEL; i += 1024) {
    topU[b * KSEL + i] = 0u; topIdx[b * KSEL + i] = -1;
  }
}

// ---------------- k4: in-LDS bitonic sort (desc) + gather boxes/labels ----------------
__global__ void __launch_bounds__(1024)
k_sortgather(const unsigned* __restrict__ topU, const int* __restrict__ topIdx,
             const float* __restrict__ boxes, const float* __restrict__ cls,
             float4* __restrict__ sBox, float* __restrict__ sScore, int* __restrict__ sLabel) {
  const int b = blockIdx.x;
  const int tid = threadIdx.x;
  __shared__ unsigned key[KSEL];
  __shared__ int val[KSEL];
  for (int i = tid; i < KSEL; i += 1024) { key[i] = topU[b * KSEL + i]; val[i] = topIdx[b * KSEL + i]; }
  __syncthreads();
  for (unsigned kk = 2; kk <= KSEL; kk <<= 1) {
    for (unsigned j = kk >> 1; j > 0; j >>= 1) {
      for (unsigned i = tid; i < KSEL; i += 1024) {
        unsigned p = i ^ j;
        if (p > i) {
          unsigned ki = key[i], kp = key[p];
          int vi = val[i], vp = val[p];
          bool desc = ((i & kk) == 0);
          bool iLess = (ki < kp) || (ki == kp && vi > vp);  // tie-break: smaller idx first
          bool iGreater = (ki > kp) || (ki == kp && vi < vp);
          if (desc ? iLess : iGreater) {
            key[i] = kp; key[p] = ki; val[i] = vp; val[p] = vi;
          }
        }
      }
      __syncthreads();
    }
  }
  for (int i = tid; i < KSEL; i += 1024) {
    unsigned u = key[i]; int idx = val[i];
    float4 bx = {0.f, 0.f, 0.f, 0.f};
    float sc = -__builtin_huge_valf();
    int lb = 0;
    if (u != 0u && idx >= 0) {
      const float* bp = boxes + ((size_t)b * NBOX + idx) * 4;
      bx.x = bp[0]; bx.y = bp[1]; bx.z = bp[2]; bx.w = bp[3];
      sc = fromOrderable(u);
      const float* cp = cls + ((size_t)b * NBOX + idx) * NCLS;
      float m = cp[0];
#pragma unroll
      for (int c = 1; c < NCLS; ++c) { if (cp[c] > m) { m = cp[c]; lb = c; } }
    }
    sBox[b * KSEL + i] = bx; sScore[b * KSEL + i] = sc; sLabel[b * KSEL + i] = lb;
  }
}

// ---------------- k5: IoU suppression bitmask ----------------
// Block = 256 thr (8 waves) -> tile of 128 rows x 32 cols (one mask word / row).
// Boxes staged via CDNA5 async-to-LDS; pairwise area-sum via V_WMMA_F32_16X16X4_F32
// (rank-2 outer sum: C[m][n] = area_row[m]*1 + 1*area_col[n]).
__global__ void __launch_bounds__(256)
k_mask(const float4* __restrict__ sBox, unsigned* __restrict__ mask) {
  const int b = blockIdx.z;
  const int colBase = blockIdx.x * 32;
  const int rowBase = blockIdx.y * 128;
  const int tid = threadIdx.x;
  __shared__ float4 cBox[32];
  __shared__ float4 rBox[128];

  if (tid < 128) {  // waves 0..3 stage row tile; wave 0 also stages column tile
    const float4* gr = sBox + (size_t)b * KSEL + rowBase + tid;
    unsigned lr = (unsigned)(unsigned long long)&rBox[tid];   // generic LDS addr: low 32 bits = LDS offset
    asm volatile("global_load_async_to_lds_b128 %0, %1, off" :: "v"(lr), "v"(gr) : "memory");
    if (tid < 32) {
      const float4* gc = sBox + (size_t)b * KSEL + colBase + tid;
      unsigned lc = (unsigned)(unsigned long long)&cBox[tid];
      asm volatile("global_load_async_to_lds_b128 %0, %1, off" :: "v"(lc), "v"(gc) : "memory");
    }
    asm volatile("s_wait_asynccnt 0" ::: "memory");
  }
  __syncthreads();

  const int lane = tid & 31;
  const int wv = tid >> 5;
  const int rOff = wv * 16;          // 8 waves x 16 rows = 128 rows
  const int n = lane & 15;
  const int hi = lane >> 4;

  unsigned balA[8], balB[8];
#pragma unroll
  for (int t = 0; t < 2; ++t) {      // two 16-col groups
    const int cOff = t * 16;
    float4 cb = cBox[cOff + n];
    float cArea = (cb.z - cb.x) * (cb.w - cb.y);
    float4 rbn = rBox[rOff + n];
    float rArea = (rbn.z - rbn.x) * (rbn.w - rbn.y);
    // A (16x4): row m = [area_row[m], 1, 0, 0]; B (4x16): col n = [1, area_col[n], 0, 0]^T
    v2f A = { hi ? 0.f : rArea, hi ? 0.f : 1.f };
    v2f Bm = { hi ? 0.f : 1.f, hi ? 0.f : cArea };
    v8f C = {};
    C = __builtin_amdgcn_wmma_f32_16x16x4_f32(false, A, false, Bm, (short)0, C, false, false);
    const int colIdx = colBase + cOff + n;
#pragma unroll
    for (int r = 0; r < 8; ++r) {
      const int m = hi * 8 + r;                 // matches C[r] layout: M = r + 8*(lane>>4), N = lane&15
      float4 rb = rBox[rOff + m];
      float ix1 = fmaxf(rb.x, cb.x), iy1 = fmaxf(rb.y, cb.y);
      float ix2 = fminf(rb.z, cb.z), iy2 = fminf(rb.w, cb.w);
      float inter = fmaxf(ix2 - ix1, 0.f) * fmaxf(iy2 - iy1, 0.f);
      // IoU > TH  <=>  inter*(1+TH) > TH*(areaSum + eps)   (divide-free)
      bool pred = (inter * (1.f + NMS_TH) > NMS_TH * (C[r] + 1e-6f)) &&
                  (colIdx > rowBase + rOff + m);
      unsigned bm = (unsigned)__ballot(pred);
      if (t == 0) balA[r] = bm; else balB[r] = bm;
    }
  }
  // ballot bits: [15:0] = row r cols 0..15, [31:16] = row r+8 cols 0..15
#pragma unroll
  for (int r = 0; r < 8; ++r) {
    unsigned wlo = (balA[r] & 0xFFFFu) | ((balB[r] & 0xFFFFu) << 16);
    unsigned whi = ((balA[r] >> 16) & 0xFFFFu) | (balB[r] & 0xFFFF0000u);
    if (lane == r)
      mask[((size_t)b * KSEL + rowBase + rOff + r) * NWORD + blockIdx.x] = wlo;
    if (lane == r + 16)
      mask[((size_t)b * KSEL + rowBase + rOff + 8 + r) * NWORD + blockIdx.x] = whi;
  }
}

// ---------------- k6: serial greedy suppression (torchvision-style) ----------------
__global__ void __launch_bounds__(128)
k_greedy(const float* __restrict__ sScore, const unsigned* __restrict__ mask,
         int* __restrict__ keepArr, int* __restrict__ keepCnt) {
  const int b = blockIdx.x;
  const int tid = threadIdx.x;   // owns mask word `tid` of each row
  __shared__ int sAlive;
  unsigned remv = 0;
  int cnt = 0;
  for (int i = 0; i < KSEL; ++i) {
    const int w = i >> 5;
    if (tid == w) {
      bool alive = (((remv >> (i & 31)) & 1u) == 0u) &&
                   (sScore[b * KSEL + i] > -1e37f);
      sAlive = alive ? 1 : 0;
    }
    __syncthreads();
    int a = sAlive;
    if (tid == 0) { keepArr[b * KSEL + i] = a; cnt += a; }
    if (a) remv |= mask[((size_t)b * KSEL + i) * NWORD + tid];
    __syncthreads();
  }
  if (tid == 0) keepCnt[b] = cnt;
}

// ---------------- k7: ballot-compact first POST kept rows, write output ----------------
__global__ void __launch_bounds__(32)
k_final(const float4* __restrict__ sBox, const float* __restrict__ sScore,
        const int* __restrict__ sLabel, const int* __restrict__ keepArr,
        float* __restrict__ out) {
  const int b = blockIdx.x;
  const int lane = threadIdx.x;
  float* ob = out + (size_t)b * POST * 6;
  for (int i = lane; i < POST * 6; i += 32) ob[i] = 0.f;
  __syncthreads();
  int cnt = 0;
  const unsigned laneLt = (1u << lane) - 1u;
  for (int g = 0; g < KSEL / 32; ++g) {
    int i = g * 32 + lane;
    bool k = keepArr[b * KSEL + i] != 0;
    unsigned m = (unsigned)__ballot(k);
    int pos = cnt + __popc(m & laneLt);
    if (k && pos < POST) {
      float4 bx = sBox[b * KSEL + i];
      ob[pos * 6 + 0] = bx.x; ob[pos * 6 + 1] = bx.y;
      ob[pos * 6 + 2] = bx.z; ob[pos * 6 + 3] = bx.w;
      ob[pos * 6 + 4] = sScore[b * KSEL + i];
      ob[pos * 6 + 5] = (float)sLabel[b * KSEL + i];
    }
    cnt += __popc(m);
  }
  if (lane == 0) {
    int v = cnt < POST ? cnt : POST;
    ((int*)out)[BATCH * POST * 6 + b] = v;   // int32 bits of n_valid
  }
}

// ---------------- host-side launch ----------------
extern "C" void kernel_launch(void* const* d_in, const int* in_sizes, int n_in,
                              void* d_out, int out_size, void* d_ws, size_t ws_size,
                              hipStream_t stream) {
  const float* boxes = (const float*)d_in[0];   // (B,N,4)
  const float* cls   = (const float*)d_in[1];   // (B,N,C)
  float* out = (float*)d_out;                   // (B,500,6) + (B,) int bits

  char* w = (char*)d_ws;
  int*      hist    = (int*)w;       w += (size_t)BATCH * 65536 * 4;
  unsigned* uArr    = (unsigned*)w;  w += (size_t)BATCH * NBOX * 4;
  int*      params  = (int*)w;       w += 64;
  unsigned* topU    = (unsigned*)w;  w += (size_t)BATCH * KSEL * 4;
  int*      topIdx  = (int*)w;       w += (size_t)BATCH * KSEL * 4;
  float4*   sBox    = (float4*)w;    w += (size_t)BATCH * KSEL * 16;
  float*    sScore  = (float*)w;     w += (size_t)BATCH * KSEL * 4;
  int*      sLabel  = (int*)w;       w += (size_t)BATCH * KSEL * 4;
  unsigned* maskBuf = (unsigned*)w;  w += (size_t)BATCH * KSEL * NWORD * 4;
  int*      keepArr = (int*)w;       w += (size_t)BATCH * KSEL * 4;
  int*      keepCnt = (int*)w;       w += (size_t)BATCH * 4;

  k_zero<<<(BATCH * 65536 + 1023) / 1024, 1024, 0, stream>>>(hist, BATCH * 65536);
  k_score<<<(BATCH * NBOX + 255) / 256, 256, 0, stream>>>(cls, uArr, hist);
  k_select<<<BATCH, 1024, 0, stream>>>(hist, params);
  k_compact<<<BATCH, 1024, 0, stream>>>(uArr, params, topU, topIdx);
  k_sortgather<<<BATCH, 1024, 0, stream>>>(topU, topIdx, boxes, cls, sBox, sScore, sLabel);
  dim3 mg(KSEL / 32, KSEL / 128, BATCH);
  k_mask<<<mg, 256, 0, stream>>>(sBox, maskBuf);
  k_greedy<<<BATCH, 128, 0, stream>>>(sScore, maskBuf, keepArr, keepCnt);
  k_final<<<BATCH, 32, 0, stream>>>(sBox, sScore, sLabel, keepArr, out);
}